// Model_34084860461512
// MI455X (gfx1250) — compile-verified
//
#include <hip/hip_runtime.h>
#include <math.h>

typedef float v2f __attribute__((ext_vector_type(2)));
typedef float v8f __attribute__((ext_vector_type(8)));

#define R_ROWS 8192

__device__ __forceinline__ float siluf(float x) { return x / (1.f + __expf(-x)); }
__device__ __forceinline__ float softplusf(float x) { return (x > 20.f) ? x : log1pf(__expf(x)); }

// ---------------------------------------------------------------------------
// WMMA fp32 GEMM: C[M,N] = act(A[M,K] @ W[N,K]^T + bias) (+ res[M,N])
// One wave computes a 16(M) x 16*NSUB(N) tile via V_WMMA_F32_16X16X4_F32.
// Branch-free inner loop: B row index clamped to N-1 (garbage columns are
// computed but never stored); only the epilogue guards on n < N.
// act: 0 = none, 1 = relu, 2 = softplus
// ---------------------------------------------------------------------------
template <int NSUB>
__global__ __launch_bounds__(128)
void gemm_wmma_kernel(const float* __restrict__ A, int lda,
                      const float* __restrict__ W,
                      const float* __restrict__ bias,
                      const float* __restrict__ res, int ldres,
                      float* __restrict__ C, int ldc,
                      int M, int N, int K, int act)
{
  const int wave  = threadIdx.x >> 5;
  const int lane  = threadIdx.x & 31;
  const int m0    = (blockIdx.y * 4 + wave) * 16;
  const int n0    = blockIdx.x * (16 * NSUB);
  const int mrow  = lane & 15;
  const int khalf = (lane >> 4) << 1;   // 0 or 2

  v8f acc[NSUB];
#pragma unroll
  for (int s = 0; s < NSUB; ++s)
#pragma unroll
    for (int v = 0; v < 8; ++v) acc[s][v] = 0.f;

  // k-invariant pointers (clamped row for out-of-range columns)
  const float* ap = A + (size_t)(m0 + mrow) * lda + khalf;
  const float* wp[NSUB];
#pragma unroll
  for (int s = 0; s < NSUB; ++s) {
    int n = n0 + s * 16 + mrow;
    n = (n < N) ? n : (N - 1);
    wp[s] = W + (size_t)n * K + khalf;
  }

  for (int k = 0; k < K; k += 4) {
    const v2f af = *(const v2f*)(ap + k);
    v2f bf[NSUB];
#pragma unroll
    for (int s = 0; s < NSUB; ++s) bf[s] = *(const v2f*)(wp[s] + k);
#pragma unroll
    for (int s = 0; s < NSUB; ++s)
      acc[s] = __builtin_amdgcn_wmma_f32_16x16x4_f32(
          false, af, false, bf[s], (short)0, acc[s], false, false);
  }

  // C/D layout: VGPR v, lanes 0-15 -> M = v, lanes 16-31 -> M = v + 8
  const int mbase = m0 + ((lane >> 4) << 3);
  const int ncol  = lane & 15;
#pragma unroll
  for (int s = 0; s < NSUB; ++s) {
    const int n = n0 + s * 16 + ncol;
    if (n >= N) continue;
    const float bv = bias ? bias[n] : 0.f;
#pragma unroll
    for (int v = 0; v < 8; ++v) {
      const int m = mbase + v;
      float val = acc[s][v] + bv;
      if (act == 1)      val = fmaxf(val, 0.f);
      else if (act == 2) val = softplusf(val);
      if (res) val += res[(size_t)m * ldres + n];
      C[(size_t)m * ldc + n] = val;
    }
  }
}

// ---------------------------------------------------------------------------
// RevIN statistics: one wave per (batch, channel) row of 512 samples
// ---------------------------------------------------------------------------
__global__ __launch_bounds__(128)
void revin_stats_kernel(const float* __restrict__ x, float* __restrict__ mean,
                        float* __restrict__ stdv)
{
  const int wave = threadIdx.x >> 5;
  const int lane = threadIdx.x & 31;
  const int p = blockIdx.x * 4 + wave;            // p = b*128 + c
  if (p >= R_ROWS) return;
  const int b = p >> 7, c = p & 127;
  const float* base = x + (size_t)b * 512 * 128 + c;
  float s = 0.f, ss = 0.f;
  for (int i = lane; i < 512; i += 32) {
    float v = base[(size_t)i * 128];
    s += v; ss += v * v;
  }
  for (int o = 16; o; o >>= 1) { s += __shfl_xor(s, o, 32); ss += __shfl_xor(ss, o, 32); }
  if (lane == 0) {
    const float m = s / 512.f;
    const float var = ss / 512.f - m * m;
    mean[p] = m;
    stdv[p] = sqrtf(var + 1e-5f);
  }
}

// H0[(b*128+c)*512 + s] = (x[b,s,c] - mean)/std
__global__ void norm_transpose_kernel(const float* __restrict__ x,
                                      const float* __restrict__ mean,
                                      const float* __restrict__ stdv,
                                      float* __restrict__ H0)
{
  const int t = blockIdx.x * blockDim.x + threadIdx.x;   // b*65536 + s*128 + c
  const int c = t & 127;
  const int s = (t >> 7) & 511;
  const int b = t >> 16;
  const int r = (b << 7) | c;
  H0[(size_t)r * 512 + s] = (x[t] - mean[r]) / stdv[r];
}

// out[b,p,c] = OUTB[(b*128+c)*96 + p] * std + mean
__global__ void final_out_kernel(const float* __restrict__ OUTB,
                                 const float* __restrict__ mean,
                                 const float* __restrict__ stdv,
                                 float* __restrict__ out)
{
  const int t = blockIdx.x * blockDim.x + threadIdx.x;   // b*12288 + p*128 + c
  const int c = t & 127;
  const int p = (t >> 7) % 96;
  const int b = t / (96 * 128);
  const int r = (b << 7) | c;
  out[t] = OUTB[(size_t)r * 96 + p] * stdv[r] + mean[r];
}

// xc = silu(xs * conv_w[:,3] + conv_b)   (L==1 collapses the conv to one tap)
__global__ void convsilu_kernel(const float* __restrict__ XZ,
                                const float* __restrict__ cw,
                                const float* __restrict__ cb,
                                float* __restrict__ XC, int Di, int total)
{
  const int t = blockIdx.x * blockDim.x + threadIdx.x;
  if (t >= total) return;
  const int d = t % Di;
  const int r = t / Di;
  const float v = XZ[(size_t)r * 2 * Di + d] * cw[4 * d + 3] + cb[d];
  XC[t] = siluf(v);
}

// S[r] = dot(B_row, C_row) over d_state=16
__global__ void rowdot_kernel(const float* __restrict__ DBC, int ld, int off,
                              float* __restrict__ S, int rows)
{
  const int r = blockIdx.x * blockDim.x + threadIdx.x;
  if (r >= rows) return;
  const float* p = DBC + (size_t)r * ld + off;
  float s = 0.f;
#pragma unroll
  for (int n = 0; n < 16; ++n) s += p[n] * p[16 + n];
  S[r] = s;
}

// Y = (dt*xc*S + xc*D) * silu(z)
__global__ void ygate_kernel(const float* __restrict__ DT,
                             const float* __restrict__ XC,
                             const float* __restrict__ XZ,
                             const float* __restrict__ S,
                             const float* __restrict__ Dp,
                             float* __restrict__ Y, int Di, int total)
{
  const int t = blockIdx.x * blockDim.x + threadIdx.x;
  if (t >= total) return;
  const int d = t % Di;
  const int r = t / Di;
  const float xc = XC[t];
  const float y = DT[t] * xc * S[r] + xc * Dp[d];
  const float z = XZ[(size_t)r * 2 * Di + Di + d];
  Y[t] = y * siluf(z);
}

// out = LN(a (+ b)) * g + beta ; one wave per row of width N (N <= 256)
__global__ __launch_bounds__(128)
void lnres_kernel(const float* __restrict__ a, const float* __restrict__ b,
                  const float* __restrict__ g, const float* __restrict__ be,
                  float* __restrict__ out, int N, int rows)
{
  const int wave = threadIdx.x >> 5;
  const int lane = threadIdx.x & 31;
  const int r = blockIdx.x * 4 + wave;
  if (r >= rows) return;
  const float* ar = a + (size_t)r * N;
  const float* br = b ? b + (size_t)r * N : nullptr;
  float vals[8];
  const int per = N / 32;
  float sum = 0.f;
  for (int i = 0; i < per; ++i) {
    float v = ar[lane + i * 32];
    if (br) v += br[lane + i * 32];
    vals[i] = v; sum += v;
  }
  for (int o = 16; o; o >>= 1) sum += __shfl_xor(sum, o, 32);
  const float mean = sum / (float)N;
  float vs = 0.f;
  for (int i = 0; i < per; ++i) { const float d = vals[i] - mean; vs += d * d; }
  for (int o = 16; o; o >>= 1) vs += __shfl_xor(vs, o, 32);
  const float rstd = rsqrtf(vs / (float)N + 1e-5f);
  float* orow = out + (size_t)r * N;
  for (int i = 0; i < per; ++i) {
    const int idx = lane + i * 32;
    orow[idx] = (vals[i] - mean) * rstd * g[idx] + be[idx];
  }
}

__global__ void add_kernel(const float* __restrict__ a, const float* __restrict__ b,
                           float* __restrict__ c, int n)
{
  const int t = blockIdx.x * blockDim.x + threadIdx.x;
  if (t < n) c[t] = a[t] + b[t];
}

// ---------------------------------------------------------------------------
// mamba4: true selective scan; d_model=1, d_inner=2, d_state=16, L=512.
// One thread per row. Writes HC[:, 512 + l] = X3[r,l] + mamba4_out.
// ---------------------------------------------------------------------------
__global__ __launch_bounds__(128)
void mamba4_scan_kernel(const float* __restrict__ H1, const float* __restrict__ X3,
                        float* __restrict__ HC,
                        const float* __restrict__ A_log, const float* __restrict__ Dp,
                        const float* __restrict__ conv_b, const float* __restrict__ conv_w,
                        const float* __restrict__ dt_b, const float* __restrict__ dt_w,
                        const float* __restrict__ in_w, const float* __restrict__ out_w,
                        const float* __restrict__ x_w)
{
  const int r = blockIdx.x * blockDim.x + threadIdx.x;
  if (r >= R_ROWS) return;

  float A[2][16];
#pragma unroll
  for (int d = 0; d < 2; ++d)
#pragma unroll
    for (int n = 0; n < 16; ++n) A[d][n] = -__expf(A_log[d * 16 + n]);
  float xw[33][2];
#pragma unroll
  for (int j = 0; j < 33; ++j) { xw[j][0] = x_w[2 * j]; xw[j][1] = x_w[2 * j + 1]; }
  float cw[2][4];
#pragma unroll
  for (int d = 0; d < 2; ++d)
#pragma unroll
    for (int k = 0; k < 4; ++k) cw[d][k] = conv_w[d * 4 + k];
  const float cb0 = conv_b[0], cb1 = conv_b[1];
  const float dtw0 = dt_w[0], dtw1 = dt_w[1];
  const float dtb0 = dt_b[0], dtb1 = dt_b[1];
  const float w0 = in_w[0], w1 = in_w[1], w2 = in_w[2], w3 = in_w[3];
  const float D0 = Dp[0], D1 = Dp[1];
  const float ow0 = out_w[0], ow1 = out_w[1];

  float h[2][16];
#pragma unroll
  for (int d = 0; d < 2; ++d)
#pragma unroll
    for (int n = 0; n < 16; ++n) h[d][n] = 0.f;
  float h0a = 0.f, h0b = 0.f, h0c = 0.f, h0d = 0.f;   // conv history ch0
  float h1a = 0.f, h1b = 0.f, h1c = 0.f, h1d = 0.f;   // conv history ch1

  const float* row   = H1 + (size_t)r * 512;
  const float* x3row = X3 + (size_t)r * 512;
  float* hcrow = HC + (size_t)r * 1024 + 512;

  for (int l = 0; l < 512; ++l) {
    const float u = row[l];
    const float xs0 = u * w0, xs1 = u * w1, z0 = u * w2, z1 = u * w3;
    h0a = h0b; h0b = h0c; h0c = h0d; h0d = xs0;
    h1a = h1b; h1b = h1c; h1c = h1d; h1d = xs1;
    float xc0 = h0a * cw[0][0] + h0b * cw[0][1] + h0c * cw[0][2] + h0d * cw[0][3] + cb0;
    float xc1 = h1a * cw[1][0] + h1b * cw[1][1] + h1c * cw[1][2] + h1d * cw[1][3] + cb1;
    xc0 = siluf(xc0);
    xc1 = siluf(xc1);
    const float dtr = xc0 * xw[0][0] + xc1 * xw[0][1];
    const float dt0 = softplusf(dtr * dtw0 + dtb0);
    const float dt1 = softplusf(dtr * dtw1 + dtb1);
    float y0 = 0.f, y1 = 0.f;
#pragma unroll
    for (int n = 0; n < 16; ++n) {
      const float Bn = xc0 * xw[1 + n][0] + xc1 * xw[1 + n][1];
      const float Cn = xc0 * xw[17 + n][0] + xc1 * xw[17 + n][1];
      h[0][n] = h[0][n] * __expf(dt0 * A[0][n]) + dt0 * xc0 * Bn;
      h[1][n] = h[1][n] * __expf(dt1 * A[1][n]) + dt1 * xc1 * Bn;
      y0 += h[0][n] * Cn;
      y1 += h[1][n] * Cn;
    }
    y0 = (y0 + xc0 * D0) * siluf(z0);
    y1 = (y1 + xc1 * D1) * siluf(z1);
    hcrow[l] = x3row[l] + (y0 * ow0 + y1 * ow1);
  }
}

// ---------------------------------------------------------------------------
// Host-side helpers
// ---------------------------------------------------------------------------
static inline void gemm(hipStream_t st, const float* A, int lda, const float* W,
                        const float* bias, const float* res, int ldres,
                        float* C, int ldc, int M, int N, int K, int act)
{
  if (N >= 128) {
    dim3 grid((N + 127) / 128, M / 64);
    gemm_wmma_kernel<8><<<grid, dim3(128), 0, st>>>(A, lda, W, bias, res, ldres,
                                                    C, ldc, M, N, K, act);
  } else {
    dim3 grid((N + 63) / 64, M / 64);
    gemm_wmma_kernel<4><<<grid, dim3(128), 0, st>>>(A, lda, W, bias, res, ldres,
                                                    C, ldc, M, N, K, act);
  }
}

struct MambaBufs { float *XZ, *XC, *DBC, *DT, *S, *Y; };

// L==1 mamba block (closed form; A_log unused since h0 = 0)
static void mamba_l1(hipStream_t st, const float* X, float* OUT,
                     int dmodel, int dinner, int dtrank,
                     const float* inw, const float* cw, const float* cb,
                     const float* xw, const float* dtw, const float* dtb,
                     const float* Dp, const float* ow, const MambaBufs& b)
{
  const int R = R_ROWS;
  const int nbc = dtrank + 32;                       // dt_rank + 2*d_state
  gemm(st, X, dmodel, inw, nullptr, nullptr, 0, b.XZ, 2 * dinner, R, 2 * dinner, dmodel, 0);
  {
    const int total = R * dinner;
    convsilu_kernel<<<dim3((total + 255) / 256), dim3(256), 0, st>>>(b.XZ, cw, cb, b.XC, dinner, total);
  }
  gemm(st, b.XC, dinner, xw, nullptr, nullptr, 0, b.DBC, nbc, R, nbc, dinner, 0);
  gemm(st, b.DBC, nbc, dtw, dtb, nullptr, 0, b.DT, dinner, R, dinner, dtrank, 2 /*softplus*/);
  rowdot_kernel<<<dim3((R + 255) / 256), dim3(256), 0, st>>>(b.DBC, nbc, dtrank, b.S, R);
  {
    const int total = R * dinner;
    ygate_kernel<<<dim3((total + 255) / 256), dim3(256), 0, st>>>(b.DT, b.XC, b.XZ, b.S, Dp, b.Y, dinner, total);
  }
  gemm(st, b.Y, dinner, ow, nullptr, nullptr, 0, OUT, dmodel, R, dmodel, dinner, 0);
}

extern "C" void kernel_launch(void* const* d_in, const int* in_sizes, int n_in,
                              void* d_out, int out_size, void* d_ws, size_t ws_size,
                              hipStream_t stream)
{
  (void)in_sizes; (void)n_in; (void)out_size; (void)ws_size;
  auto F = [&](int i) { return (const float*)d_in[i]; };
  const int R = R_ROWS;

  // --- input pytree indices (sorted dict keys; params before x) ---
  // per-layer (base = l*28): 0 b1, 1 b2, 2 lnb_b, 3 lnb_g, 4 lnf_b, 5 lnf_g,
  //   6 lnn_b, 7 lnn_g, 8..16 mb{A_log,D,conv_b,conv_w,dt_b,dt_w,in_w,out_w,x_w},
  //   17..25 mf{...same...}, 26 w1, 27 w2
  // globals: 56 lin1_b, 57 lin1_w, 58 lin2_b, 59 lin2_w, 60 lin3_b, 61 lin3_w,
  //   62 lin4_b, 63 lin4_w, 64..72 mamba3{...}, 73..81 mamba4{...},
  //   82 norm_b, 83 norm_g, 84 x
  const float* x_in = F(84);

  // --- workspace layout (floats), with region reuse ---
  float* ws = (float*)d_ws;
  size_t o = 0;
  auto alloc = [&](size_t n) { float* p = ws + o; o += n; return p; };
  float* MEAN = alloc(R);
  float* STD  = alloc(R);
  float* H1   = alloc((size_t)R * 512);    // x_res1, persists to lin3
  float* H0   = alloc((size_t)R * 512);    // input to lin1; reused as X3
  float* AR1  = alloc((size_t)R * 2048);   // XZ3 | {XZe, XCe, DTe}
  float* AR2  = alloc((size_t)R * 1024);   // XC3 | {Ye, MO, F}
  float* DBCb = alloc((size_t)R * 64);     // DBC3 | DBCe
  float* AR3  = alloc((size_t)R * 1024);   // DT3 | FF1
  float* AR4  = alloc((size_t)R * 1024);   // Y3 | {B, O, FF2, EN}
  float* Sb   = alloc(R);
  float* H2   = alloc((size_t)R * 256);    // x_res2
  float* HC   = alloc((size_t)R * 1024);   // concat buffer for lin4
  float* E    = alloc((size_t)R * 256);    // encoder state; reused as H3
  float* OUTB = alloc((size_t)R * 96);

  float* X3 = H0;

  // 1) RevIN stats + normalize/transpose into H0 (8192 x 512)
  revin_stats_kernel<<<dim3(R / 4), dim3(128), 0, stream>>>(x_in, MEAN, STD);
  norm_transpose_kernel<<<dim3((64 * 512 * 128) / 256), dim3(256), 0, stream>>>(x_in, MEAN, STD, H0);

  // 2) lin1: H1 = H0 @ lin1_w^T + lin1_b
  gemm(stream, H0, 512, F(57), F(56), nullptr, 0, H1, 512, R, 512, 512, 0);

  // 3) mamba3 (L=1, d_model=512, d_inner=1024, dt_rank=32) -> X3
  {
    MambaBufs b3{AR1, AR2, DBCb, AR3, Sb, AR4};
    mamba_l1(stream, H1, X3, 512, 1024, 32,
             F(70), F(67), F(66), F(72), F(69), F(68), F(65), F(71), b3);
  }

  // 4) mamba4 scan over L=512 -> HC[:, 512:1024] = X3 + mamba4(H1)
  mamba4_scan_kernel<<<dim3(R / 128), dim3(128), 0, stream>>>(
      H1, X3, HC, F(73), F(74), F(75), F(76), F(77), F(78), F(79), F(80), F(81));

  // 5) lin2: H2 = H1 @ lin2_w^T + lin2_b  (x_res2)
  gemm(stream, H1, 512, F(59), F(58), nullptr, 0, H2, 256, R, 256, 512, 0);

  // 6) encoder layers
  float* XZe  = AR1;
  float* XCe  = AR1 + (size_t)R * 1024;
  float* DTe  = AR1 + (size_t)R * 1536;
  float* Ye   = AR2;
  float* MO   = AR2 + (size_t)R * 512;
  float* Fbuf = AR2 + (size_t)R * 768;
  float* FF1  = AR3;
  float* Bbuf = AR4;
  float* Obuf = AR4 + (size_t)R * 256;
  float* FF2  = AR4 + (size_t)R * 512;
  float* EN   = AR4 + (size_t)R * 768;
  MambaBufs be{XZe, XCe, DBCb, DTe, Sb, Ye};

  for (int l = 0; l < 2; ++l) {
    const int LB = l * 28;
    const float* ein = (l == 0) ? H2 : E;
    // forward mamba (mf) + LN
    mamba_l1(stream, ein, MO, 256, 512, 16,
             F(LB + 23), F(LB + 20), F(LB + 19), F(LB + 25),
             F(LB + 22), F(LB + 21), F(LB + 18), F(LB + 24), be);
    lnres_kernel<<<dim3(R / 4), dim3(128), 0, stream>>>(MO, ein, F(LB + 5), F(LB + 4), Fbuf, 256, R);
    // backward mamba (mb): L==1, reversal is identity
    mamba_l1(stream, ein, MO, 256, 512, 16,
             F(LB + 14), F(LB + 11), F(LB + 10), F(LB + 16),
             F(LB + 13), F(LB + 12), F(LB + 9), F(LB + 15), be);
    lnres_kernel<<<dim3(R / 4), dim3(128), 0, stream>>>(MO, ein, F(LB + 3), F(LB + 2), Bbuf, 256, R);
    // o = f + b
    add_kernel<<<dim3((R * 256) / 256), dim3(256), 0, stream>>>(Fbuf, Bbuf, Obuf, R * 256);
    // FFN
    gemm(stream, Obuf, 256, F(LB + 26), F(LB + 0), nullptr, 0, FF1, 1024, R, 1024, 256, 1 /*relu*/);
    gemm(stream, FF1, 1024, F(LB + 27), F(LB + 1), nullptr, 0, FF2, 256, R, 256, 1024, 0);
    lnres_kernel<<<dim3(R / 4), dim3(128), 0, stream>>>(FF2, Obuf, F(LB + 7), F(LB + 6), E, 256, R);
  }

  // 7) final norm, h3 = x_res2 + ln(e)
  lnres_kernel<<<dim3(R / 4), dim3(128), 0, stream>>>(E, nullptr, F(83), F(82), EN, 256, R);
  add_kernel<<<dim3((R * 256) / 256), dim3(256), 0, stream>>>(H2, EN, E /*H3*/, R * 256);

  // 8) lin3 (+x_res1) into HC[:, 0:512]
  gemm(stream, E, 256, F(61), F(60), H1, 512, HC, 1024, R, 512, 256, 0);

  // 9) lin4 over concat (K=1024) -> OUTB (8192 x 96)
  gemm(stream, HC, 1024, F(63), F(62), nullptr, 0, OUTB, 96, R, 96, 1024, 0);

  // 10) de-normalize + transpose to (64, 96, 128)
  final_out_kernel<<<dim3((64 * 96 * 128) / 256), dim3(256), 0, stream>>>(
      OUTB, MEAN, STD, (float*)d_out);
}